// LSTM_40364102648410
// MI455X (gfx1250) — compile-verified
//
#include <hip/hip_runtime.h>
#include <hip/hip_bf16.h>

typedef __attribute__((ext_vector_type(2))) float v2f;
typedef __attribute__((ext_vector_type(8))) float v8f;

#define SEQ_T 32768
#define IN_D  512
#define HID_H 20
#define G4H   80   // 4*H

// ---------------------------------------------------------------------------
// Kernel 1: xg[T][80] = x[T][512] @ W_ih^T + (b_ih + b_hh)
// One wave32 per 16-row M tile; 5 N-tiles of 16 (80 = 5*16); K consumed 4 at a
// time with V_WMMA_F32_16X16X4_F32 (full fp32 — GEMM is HBM-bound so there is
// no reason to drop precision for matrix-core rate).
//
// Fragment layouts (CDNA5 ISA 7.12.2, wave32):
//   A 16x4 f32 : VGPR j, lanes 0-15 -> (M=lane,   K = j + 2*0)
//                         lanes16-31 -> (M=lane-16,K = j + 2*1)
//   B 4x16 f32 : VGPR j, lanes 0-15 -> (K = j,     N=lane)
//                         lanes16-31 -> (K = j + 2, N=lane-16)
//   C/D 16x16  : VGPR v, lanes 0-15 -> (M=v,   N=lane)
//                         lanes16-31 -> (M=v+8, N=lane-16)
// ---------------------------------------------------------------------------
__global__ __launch_bounds__(32) void lstm_gates_gemm(
    const float* __restrict__ x,     // (T, D)
    const float* __restrict__ W_ih,  // (80, 512) row-major
    const float* __restrict__ b_ih,  // (80,)
    const float* __restrict__ b_hh,  // (80,)
    float* __restrict__ xg)          // (T, 80)
{
    const int mtile = blockIdx.x;        // 0..T/16-1
    const int lane  = threadIdx.x;       // 0..31
    const int half  = lane >> 4;         // 0 | 1  -> K offset 0 | 2
    const int l15   = lane & 15;

    // Accumulators preloaded with bias (bias depends only on N = l15 column).
    v8f acc[5];
#pragma unroll
    for (int nt = 0; nt < 5; ++nt) {
        const int n = nt * 16 + l15;
        const float bias = b_ih[n] + b_hh[n];
#pragma unroll
        for (int v = 0; v < 8; ++v) acc[nt][v] = bias;
    }

    const float* xrow = x + (size_t)(mtile * 16 + l15) * IN_D;

    for (int k0 = 0; k0 < IN_D; k0 += 4) {
        // A fragment: two consecutive K values per lane -> one b64 load.
        v2f a = *(const v2f*)(xrow + k0 + 2 * half);
#pragma unroll
        for (int nt = 0; nt < 5; ++nt) {
            // B[k][n] = W_ih[n][k]; K contiguous in W_ih row -> one b64 load.
            v2f b = *(const v2f*)(W_ih + (size_t)(nt * 16 + l15) * IN_D + k0 + 2 * half);
            acc[nt] = __builtin_amdgcn_wmma_f32_16x16x4_f32(
                /*neg_a=*/false, a, /*neg_b=*/false, b,
                /*c_mod=*/(short)0, acc[nt],
                /*reuse_a=*/false, /*reuse_b=*/false);
        }
    }

    // Store D: element (M = v + 8*half, N = l15) of tile.
    float* outbase = xg + (size_t)(mtile * 16) * G4H;
#pragma unroll
    for (int nt = 0; nt < 5; ++nt) {
#pragma unroll
        for (int v = 0; v < 8; ++v) {
            outbase[(size_t)(v + 8 * half) * G4H + nt * 16 + l15] = acc[nt][v];
        }
    }
}

// ---------------------------------------------------------------------------
// Kernel 2: the serial recurrence. Single wave32; lane n (n<20) owns hidden
// unit n and its four W_hh gate rows in VGPRs. h is broadcast each step with
// v_readlane into uniform (SGPR) values so every recurrent MAC is
// v_fmac_f32 vdst, sgpr, vgpr. xg for t+1 is prefetched during step t.
// The output projection (W_out . h + b_out) is fused here (h is uniform after
// the broadcast), so hs is never materialized.
// ---------------------------------------------------------------------------
__device__ __forceinline__ float readlane_f(float v, int l) {
    return __builtin_bit_cast(float,
        __builtin_amdgcn_readlane(__builtin_bit_cast(int, v), l));
}

__device__ __forceinline__ float sigmoid_f(float v) {
    return 1.0f / (1.0f + __expf(-v));
}

__global__ __launch_bounds__(32) void lstm_scan(
    const float* __restrict__ xg,     // (T, 80)
    const float* __restrict__ W_hh,   // (80, 20)
    const float* __restrict__ W_out,  // (1, 20)
    const float* __restrict__ b_out,  // (1,)
    const float* __restrict__ h0,     // (20,)
    const float* __restrict__ c0,     // (20,)
    float* __restrict__ out)          // [T | hT(20) | cT(20)]
{
    const int lane   = threadIdx.x;
    const bool active = lane < HID_H;
    const int  n      = active ? lane : 0;   // inactive lanes shadow unit 0

    // Per-lane recurrent weight rows (gate order i, f, g, o).
    float wi[HID_H], wf[HID_H], wg[HID_H], wo[HID_H], wout[HID_H];
#pragma unroll
    for (int k = 0; k < HID_H; ++k) {
        wi[k]   = W_hh[(size_t)(0 * HID_H + n) * HID_H + k];
        wf[k]   = W_hh[(size_t)(1 * HID_H + n) * HID_H + k];
        wg[k]   = W_hh[(size_t)(2 * HID_H + n) * HID_H + k];
        wo[k]   = W_hh[(size_t)(3 * HID_H + n) * HID_H + k];
        wout[k] = W_out[k];
    }
    const float bout = b_out[0];

    float c = c0[n];
    float h = h0[n];

    // Uniform broadcast of h.
    float hb[HID_H];
#pragma unroll
    for (int k = 0; k < HID_H; ++k) hb[k] = readlane_f(h, k);

    // Prime the xg pipeline for t = 0.
    float xi = xg[0 * G4H + 0 * HID_H + n];
    float xf = xg[0 * G4H + 1 * HID_H + n];
    float xc = xg[0 * G4H + 2 * HID_H + n];
    float xo = xg[0 * G4H + 3 * HID_H + n];

    for (int t = 0; t < SEQ_T; ++t) {
        // Prefetch next step's gate pre-activations while we compute.
        float nxi = 0.f, nxf = 0.f, nxc = 0.f, nxo = 0.f;
        if (t + 1 < SEQ_T) {
            const float* p = xg + (size_t)(t + 1) * G4H;
            nxi = p[0 * HID_H + n];
            nxf = p[1 * HID_H + n];
            nxc = p[2 * HID_H + n];
            nxo = p[3 * HID_H + n];
        }

        float gi = xi, gf = xf, gc = xc, go = xo;
#pragma unroll
        for (int k = 0; k < HID_H; ++k) {
            gi = fmaf(wi[k], hb[k], gi);
            gf = fmaf(wf[k], hb[k], gf);
            gc = fmaf(wg[k], hb[k], gc);
            go = fmaf(wo[k], hb[k], go);
        }

        const float i_ = sigmoid_f(gi);
        const float f_ = sigmoid_f(gf);
        const float o_ = sigmoid_f(go);
        const float g_ = tanhf(gc);

        c = fmaf(f_, c, i_ * g_);
        h = o_ * tanhf(c);

        // Broadcast new h to uniform registers for the next matvec + output.
#pragma unroll
        for (int k = 0; k < HID_H; ++k) hb[k] = readlane_f(h, k);

        // Fused output projection out[t] = b_out + W_out . h  (uniform math).
        float ot = bout;
#pragma unroll
        for (int k = 0; k < HID_H; ++k) ot = fmaf(wout[k], hb[k], ot);
        if (lane == 0) out[t] = ot;

        xi = nxi; xf = nxf; xc = nxc; xo = nxo;
    }

    if (active) {
        out[SEQ_T + n]         = h;   // hT
        out[SEQ_T + HID_H + n] = c;   // cT
    }
}

// ---------------------------------------------------------------------------
// setup_inputs() order:
//   0:x 1:h_state 2:c_state 3:W_ih 4:W_hh 5:b_ih 6:b_hh 7:W_out 8:b_out
// d_out: outputs(T) | hT(20) | cT(20)   (all fp32)
// d_ws: xg scratch, T*80*4 = 10.5 MB
// ---------------------------------------------------------------------------
extern "C" void kernel_launch(void* const* d_in, const int* in_sizes, int n_in,
                              void* d_out, int out_size, void* d_ws, size_t ws_size,
                              hipStream_t stream) {
    (void)in_sizes; (void)n_in; (void)out_size; (void)ws_size;

    const float* x    = (const float*)d_in[0];
    const float* h0   = (const float*)d_in[1];
    const float* c0   = (const float*)d_in[2];
    const float* W_ih = (const float*)d_in[3];
    const float* W_hh = (const float*)d_in[4];
    const float* b_ih = (const float*)d_in[5];
    const float* b_hh = (const float*)d_in[6];
    const float* W_out= (const float*)d_in[7];
    const float* b_out= (const float*)d_in[8];

    float* out = (float*)d_out;
    float* xg  = (float*)d_ws;   // (T, 80)

    // Phase 1: bandwidth-bound WMMA GEMM, one wave per 16-row tile.
    lstm_gates_gemm<<<SEQ_T / 16, 32, 0, stream>>>(x, W_ih, b_ih, b_hh, xg);

    // Phase 2: latency-bound serial scan, single wave.
    lstm_scan<<<1, 32, 0, stream>>>(xg, W_hh, W_out, b_out, h0, c0, out);
}